// TextEncoder_11020886082261
// MI455X (gfx1250) — compile-verified
//
#include <hip/hip_runtime.h>
#include <cstdint>
#include <cstddef>

// ---------------------------------------------------------------------------
// 2-layer LSTM text encoder for gfx1250 (MI455X), wave32 + V_WMMA_F32_16X16X4.
//
// Design: recurrence is the serial bottleneck (1024 dependent steps). Batch
// rows are independent across the recurrence, so 8 workgroups each own a
// 16-row batch slice and run all T steps internally: h state in LDS,
// c state resident in WMMA accumulator registers (layout is step-invariant).
// Each wave (16 waves/block) owns 32 gate columns => 8 accum tiles
// (i,f,g,o gates x 2 N-tiles); gate nonlinearity is per-lane VALU since the
// i/f/g/o tiles share the identical 16x16 C-layout.
//
// Workspace: hs1 [T,B,H] f32 = 134,217,728 B, pooled [B,H] = 262,144 B.
// ---------------------------------------------------------------------------

typedef float v2f __attribute__((ext_vector_type(2)));
typedef float v8f __attribute__((ext_vector_type(8)));

#define HID   512
#define SEQT  512
#define BATCH 128
#define VOCABN 1000
#define LDSW  516            // padded LDS row stride (floats): conflict-free A loads

#define OFF_HID  128000      // encoded[128,1000] | hidden[2,128,512] | cell[2,128,512]
#define OFF_CELL 259072

__device__ __forceinline__ float sigm(float x) { return 1.0f / (1.0f + __expf(-x)); }

template<int LAYER>
__global__ __launch_bounds__(512)
void lstm_layer_kernel(const int*   __restrict__ tokens,  // [B,T]        (LAYER==0)
                       const float* __restrict__ Wx,      // L0: [2048,1000] gather; L1: [2048,512] B-matrix
                       const float* __restrict__ Whh,     // [2048,512]
                       const float* __restrict__ b_ih,    // [2048]
                       const float* __restrict__ b_hh,    // [2048]
                       const float* __restrict__ xseq,    // L1: hs1 [T,B,H]
                       float*       __restrict__ hs_out,  // L0: hs1 [T,B,H]
                       float*       __restrict__ pooled,  // L1: [B,H]
                       float*       __restrict__ out)     // d_out base
{
  __shared__ float hbuf[16 * LDSW];

  const int lane  = threadIdx.x & 31;
  const int wave  = threadIdx.x >> 5;   // 0..15
  const int lhalf = lane >> 4;          // half-wave select (ISA tile layout)
  const int lcol  = lane & 15;
  const int nb    = wave * 32;          // this wave's 2 N-tiles: nb, nb+16
  const int bs    = blockIdx.x * 16;    // batch slice base row

  // h0 = 0
  for (int i = threadIdx.x; i < 16 * LDSW; i += 512) hbuf[i] = 0.0f;

  // Per-lane gate-column indices + fused biases (t-invariant).
  // Tile tt: gate = tt>>1 ... wait; mapping: gate=(tt>>1) is wrong for i,f,g,o pairs?
  // tt = gate*2 + nt  -> gate = tt>>1 in {0,1,2,3} = {i,f,g,o}, nt = tt&1.
  int   ng[8];
  float bias[8];
  #pragma unroll
  for (int tt = 0; tt < 8; ++tt) {
    ng[tt]   = (tt >> 1) * HID + nb + (tt & 1) * 16 + lcol;
    bias[tt] = b_ih[ng[tt]] + b_hh[ng[tt]];
  }

  v8f cst[2];   // cell state, persists across all T steps in registers
  v8f pool[2];  // layer-1 time-pooled h
  #pragma unroll
  for (int e = 0; e < 8; ++e) { cst[0][e]=0.f; cst[1][e]=0.f; pool[0][e]=0.f; pool[1][e]=0.f; }

  __syncthreads();

  for (int t = 0; t < SEQT; ++t) {
    // ---- accumulator init: bias (+ embedding gather for layer 0) ----------
    v8f acc[8];
    if (LAYER == 0) {
      int tok[8];
      #pragma unroll
      for (int e = 0; e < 8; ++e)
        tok[e] = tokens[(bs + e + 8 * lhalf) * SEQT + t];
      #pragma unroll
      for (int tt = 0; tt < 8; ++tt) {
        const float* colp = Wx + (size_t)ng[tt] * VOCABN;
        #pragma unroll
        for (int e = 0; e < 8; ++e)
          acc[tt][e] = bias[tt] + colp[tok[e]];   // one-hot @ W_ih.T == column gather
      }
    } else {
      #pragma unroll
      for (int tt = 0; tt < 8; ++tt)
        #pragma unroll
        for (int e = 0; e < 8; ++e)
          acc[tt][e] = bias[tt];
    }

    const float* xs_t = (LAYER == 1) ? (xseq + ((size_t)t * BATCH + bs) * HID) : nullptr;

    // ---- gates += h_prev @ Whh.T  (+ hs1_t @ W_ih_l1.T for layer 1) -------
    // A 16x4 f32: lane<16 -> M=lane, K={kb,kb+1}; lane>=16 -> K={kb+2,kb+3}
    // B 4x16 f32: vgpr i, lane l -> K = kb + i + 2*(l>>4), N = l&15
    for (int kb = 0; kb < HID; kb += 4) {
      const int kk = kb + 2 * lhalf;
      v2f a1 = *(const v2f*)&hbuf[lcol * LDSW + kk];
      #pragma unroll
      for (int tt = 0; tt < 8; ++tt) {
        v2f bm = *(const v2f*)&Whh[(size_t)ng[tt] * HID + kk];
        acc[tt] = __builtin_amdgcn_wmma_f32_16x16x4_f32(
            false, a1, false, bm, (short)0, acc[tt], false, false);
      }
      if (LAYER == 1) {
        v2f a2 = *(const v2f*)&xs_t[lcol * HID + kk];
        #pragma unroll
        for (int tt = 0; tt < 8; ++tt) {
          v2f bm = *(const v2f*)&Wx[(size_t)ng[tt] * HID + kk];
          acc[tt] = __builtin_amdgcn_wmma_f32_16x16x4_f32(
              false, a2, false, bm, (short)0, acc[tt], false, false);
        }
      }
    }

    __syncthreads();   // all LDS h reads complete before overwriting

    // ---- gate nonlinearity + state update (pure per-lane, i/f/g/o aligned)
    #pragma unroll
    for (int nt = 0; nt < 2; ++nt) {
      #pragma unroll
      for (int e = 0; e < 8; ++e) {
        float gi = sigm(acc[0 + nt][e]);
        float gf = sigm(acc[2 + nt][e]);
        float gg = tanhf(acc[4 + nt][e]);
        float go = sigm(acc[6 + nt][e]);
        float c  = gf * cst[nt][e] + gi * gg;
        cst[nt][e] = c;
        float h  = go * tanhf(c);
        const int row = e + 8 * lhalf;            // local batch row in slice
        const int col = nb + nt * 16 + lcol;      // hidden index
        hbuf[row * LDSW + col] = h;
        if (LAYER == 0)
          hs_out[((size_t)t * BATCH + bs + row) * HID + col] = h;
        else
          pool[nt][e] += h;
        if (t == SEQT - 1)
          out[OFF_HID + LAYER * BATCH * HID + (bs + row) * HID + col] = h;
      }
    }
    __syncthreads();   // h_t visible to whole block before next step's GEMM
  }

  // ---- final cell state + pooled --------------------------------------------
  #pragma unroll
  for (int nt = 0; nt < 2; ++nt)
    #pragma unroll
    for (int e = 0; e < 8; ++e) {
      const int row = bs + e + 8 * lhalf;
      const int col = nb + nt * 16 + lcol;
      out[OFF_CELL + LAYER * BATCH * HID + row * HID + col] = cst[nt][e];
      if (LAYER == 1) pooled[row * HID + col] = pool[nt][e];
    }
}

// encoded[b][o] = pooled[b] . W_out[o] + b_out[o]   (131 MFLOP, trivial)
__global__ __launch_bounds__(256)
void out_gemm_kernel(const float* __restrict__ pooled,
                     const float* __restrict__ W_out,
                     const float* __restrict__ b_out,
                     float* __restrict__ enc)
{
  const int idx = blockIdx.x * blockDim.x + threadIdx.x;
  if (idx >= BATCH * 1000) return;
  const int b = idx / 1000;
  const int o = idx % 1000;
  const float4* pr = (const float4*)(pooled + (size_t)b * HID);
  const float4* wr = (const float4*)(W_out + (size_t)o * HID);
  float s = 0.0f;
  #pragma unroll 4
  for (int k = 0; k < HID / 4; ++k) {
    float4 p = pr[k], w = wr[k];
    s += p.x * w.x + p.y * w.y + p.z * w.z + p.w * w.w;
  }
  enc[idx] = s + b_out[o];
}

extern "C" void kernel_launch(void* const* d_in, const int* in_sizes, int n_in,
                              void* d_out, int out_size, void* d_ws, size_t ws_size,
                              hipStream_t stream) {
  const int*   tokens  = (const int*)  d_in[0];   // [128,512] int32
  const float* W_ih_l0 = (const float*)d_in[1];   // [2048,1000]
  const float* W_hh_l0 = (const float*)d_in[2];   // [2048,512]
  const float* b_ih_l0 = (const float*)d_in[3];
  const float* b_hh_l0 = (const float*)d_in[4];
  const float* W_ih_l1 = (const float*)d_in[5];   // [2048,512]
  const float* W_hh_l1 = (const float*)d_in[6];
  const float* b_ih_l1 = (const float*)d_in[7];
  const float* b_hh_l1 = (const float*)d_in[8];
  const float* W_out   = (const float*)d_in[9];   // [1000,512]
  const float* b_out   = (const float*)d_in[10];
  float* out = (float*)d_out;

  float* hs1    = (float*)d_ws;                                        // 134,217,728 B
  float* pooled = (float*)((char*)d_ws + (size_t)SEQT * BATCH * HID * sizeof(float));

  // Layer 0: embedding gather fused in-step; writes hs1, final h/c.
  lstm_layer_kernel<0><<<BATCH / 16, 512, 0, stream>>>(
      tokens, W_ih_l0, W_hh_l0, b_ih_l0, b_hh_l0,
      nullptr, hs1, nullptr, out);

  // Layer 1: input projection fused as second WMMA accumulation; pools h over T.
  lstm_layer_kernel<1><<<BATCH / 16, 512, 0, stream>>>(
      tokens, W_ih_l1, W_hh_l1, b_ih_l1, b_hh_l1,
      hs1, nullptr, pooled, out);

  out_gemm_kernel<<<(BATCH * 1000 + 255) / 256, 256, 0, stream>>>(
      pooled, W_out, b_out, out);
}